// GraphTree_26568667693191
// MI455X (gfx1250) — compile-verified
//
#include <hip/hip_runtime.h>
#include <hip/hip_bf16.h>
#include <math.h>

typedef __attribute__((ext_vector_type(2))) float v2f;
typedef __attribute__((ext_vector_type(8))) float v8f;

#define EPSC 1e-15f

__device__ __forceinline__ float block_reduce_sum(float v, float* red, int t, int nt) {
  red[t] = v; __syncthreads();
  for (int s = nt >> 1; s > 0; s >>= 1) {
    if (t < s) red[t] += red[t + s];
    __syncthreads();
  }
  float r = red[0]; __syncthreads();
  return r;
}

// ---------------------------------------------------------------------------
// Generic fp32 WMMA GEMM:  D = op(A@B + bias + Cself) with optional per-row
// mask multiply, ReLU, and fused A-row-sum output (d_flat).
// Tile 64x256, BK=32, 8 waves; each wave: 16 rows x 128 cols (8 x v8f acc).
// M % 64 == 0, K % 32 == 0; N edge-guarded. Software-pipelined global->LDS.
// ---------------------------------------------------------------------------
#define BM 64
#define BN 256
#define BK 32

__global__ __launch_bounds__(256) void wmma_gemm_f32(
    const float* __restrict__ A, const float* __restrict__ B,
    const float* __restrict__ bias, const float* __restrict__ Cself,
    const float* __restrict__ maskRow, float* __restrict__ D,
    float* __restrict__ rowsum,
    int M, int N, int K,
    long long strideA, long long strideB, long long strideD,
    int lda, int ldb, int ldd, int relu_flag)
{
  __shared__ __align__(16) float As[BM][BK + 4];    // 64 x 36
  __shared__ __align__(16) float Bs[BK][BN + 4];    // 32 x 260
  __shared__ float rs[256];

  const int b = blockIdx.z;
  const float* Ab = A + (long long)b * strideA;
  const float* Bb = B + (long long)b * strideB;
  float* Db = D + (long long)b * strideD;

  const int rowBase = blockIdx.y * BM;
  const int colBase = blockIdx.x * BN;

  const int tid  = threadIdx.x;
  const int lane = tid & 31;
  const int wave = tid >> 5;     // 0..7
  const int warpM = wave & 3;    // 0..3 -> 16-row slice
  const int warpN = wave >> 2;   // 0..1 -> 128-col slice

  const v8f vzero = {0.f, 0.f, 0.f, 0.f, 0.f, 0.f, 0.f, 0.f};
  v8f acc[8];
#pragma unroll
  for (int j = 0; j < 8; j++) acc[j] = vzero;
  float rsAcc = 0.f;

  // A staging: thread loads 8 consecutive floats of one row (2x float4)
  const int aRow = tid >> 2;           // 0..63
  const int aCol = (tid & 3) * 8;      // {0,8,16,24}
  // B staging: thread loads 32 consecutive floats of one row (8x float4)
  const int bRow = tid >> 3;           // 0..31
  const int bCol = (tid & 7) * 32;     // {0,32,...,224}

  const int frm = lane & 15;           // row (A) / col (B) within 16
  const int frh = lane >> 4;           // 0/1 -> k pair select

  float4 aReg0, aReg1;
  float4 bReg[8];

  auto loadRegs = [&](int k0) {
    const float* aPtr = Ab + (long long)(rowBase + aRow) * lda + (k0 + aCol);
    aReg0 = *(const float4*)(aPtr);
    aReg1 = *(const float4*)(aPtr + 4);
    rsAcc += aReg0.x + aReg0.y + aReg0.z + aReg0.w
           + aReg1.x + aReg1.y + aReg1.z + aReg1.w;
#pragma unroll
    for (int i = 0; i < 8; i++) {
      int gcol = colBase + bCol + 4 * i;
      const float* bPtr = Bb + (long long)(k0 + bRow) * ldb + gcol;
      float4 bv;
      if (gcol + 3 < N) {
        bv = *(const float4*)bPtr;
      } else {
        bv.x = (gcol + 0 < N) ? bPtr[0] : 0.f;
        bv.y = (gcol + 1 < N) ? bPtr[1] : 0.f;
        bv.z = (gcol + 2 < N) ? bPtr[2] : 0.f;
        bv.w = (gcol + 3 < N) ? bPtr[3] : 0.f;
      }
      bReg[i] = bv;
    }
  };
  auto storeLDS = [&]() {
    *(float4*)&As[aRow][aCol]     = aReg0;
    *(float4*)&As[aRow][aCol + 4] = aReg1;
#pragma unroll
    for (int i = 0; i < 8; i++)
      *(float4*)&Bs[bRow][bCol + 4 * i] = bReg[i];
  };

  loadRegs(0);
  storeLDS();
  for (int k0 = 0; k0 < K; k0 += BK) {
    __syncthreads();                       // LDS tiles ready
    const bool more = (k0 + BK) < K;
    if (more) loadRegs(k0 + BK);           // overlap next-chunk loads w/ WMMA
#pragma unroll
    for (int kk = 0; kk < BK; kk += 4) {
      v2f af;
      af.x = As[warpM * 16 + frm][kk + 2 * frh];
      af.y = As[warpM * 16 + frm][kk + 2 * frh + 1];
#pragma unroll
      for (int j = 0; j < 8; j++) {
        v2f bf;
        bf.x = Bs[kk + 2 * frh][warpN * 128 + j * 16 + frm];
        bf.y = Bs[kk + 2 * frh + 1][warpN * 128 + j * 16 + frm];
        acc[j] = __builtin_amdgcn_wmma_f32_16x16x4_f32(
            false, af, false, bf, (short)0, acc[j], false, false);
      }
    }
    __syncthreads();                       // compute done, LDS reusable
    if (more) storeLDS();
  }

  // epilogue: VGPR v -> M = v + 8*(lane/16), N = lane%16
#pragma unroll
  for (int j = 0; j < 8; j++) {
    int col = colBase + warpN * 128 + j * 16 + frm;
    if (col < N) {
#pragma unroll
      for (int v = 0; v < 8; v++) {
        int grow = rowBase + warpM * 16 + v + 8 * frh;
        float x = acc[j][v];
        if (bias)  x += bias[col];
        if (Cself) x += Cself[(long long)b * strideD + (long long)grow * ldd + col];
        if (relu_flag) x = fmaxf(x, 0.f);
        if (maskRow) x *= maskRow[(long long)b * M + grow];
        Db[(long long)grow * ldd + col] = x;
      }
    }
  }

  if (rowsum && blockIdx.x == 0) {
    rs[tid] = rsAcc;
    __syncthreads();
    if (tid < 64) {
      float s = rs[tid * 4] + rs[tid * 4 + 1] + rs[tid * 4 + 2] + rs[tid * 4 + 3];
      rowsum[(long long)b * M + rowBase + tid] = s;
    }
  }
}

// ---------------------------------------------------------------------------
// Row-wise L2 normalize (in place), rows of `ncols`
// ---------------------------------------------------------------------------
__global__ __launch_bounds__(128) void l2norm_rows(float* __restrict__ X, int ncols) {
  __shared__ float red[128];
  long long row = blockIdx.x;
  float* p = X + row * ncols;
  float ss = 0.f;
  for (int c = threadIdx.x; c < ncols; c += 128) { float v = p[c]; ss += v * v; }
  float tot = block_reduce_sum(ss, red, threadIdx.x, 128);
  float inv = 1.0f / (sqrtf(tot) + 1e-12f);
  for (int c = threadIdx.x; c < ncols; c += 128) p[c] *= inv;
}

// ---------------------------------------------------------------------------
// s = softmax(X @ pool_w + pool_b) * mask ; also mincut_den += d[n]*sum_k s^2
// ---------------------------------------------------------------------------
__global__ __launch_bounds__(128) void pool_softmax(
    const float* __restrict__ X, const float* __restrict__ pool_w,
    const float* __restrict__ pool_b, const float* __restrict__ mask,
    const float* __restrict__ dflat, float* __restrict__ s_out,
    float* __restrict__ mincut_den)
{
  __shared__ float xrow[512];
  __shared__ float red[128];
  long long row = blockIdx.x;          // 0..16383
  int b = (int)(row >> 13);
  const float* xp = X + row * 512;
  for (int c = threadIdx.x; c < 512; c += 128) xrow[c] = xp[c];
  __syncthreads();

  int k = threadIdx.x;
  float val = -1e30f;
  if (k < 100) {
    float a = pool_b[k];
    for (int e = 0; e < 512; e++) a += xrow[e] * pool_w[e * 100 + k];
    val = a;
  }
  red[threadIdx.x] = val; __syncthreads();
  for (int st = 64; st > 0; st >>= 1) {
    if (threadIdx.x < st) red[threadIdx.x] = fmaxf(red[threadIdx.x], red[threadIdx.x + st]);
    __syncthreads();
  }
  float mx = red[0]; __syncthreads();
  float ex = (k < 100) ? expf(val - mx) : 0.f;
  float denom = block_reduce_sum(ex, red, threadIdx.x, 128);
  float mv = mask[row];
  float sv = 0.f;
  if (k < 100) {
    sv = ex / denom * mv;
    s_out[row * 100 + k] = sv;
  }
  float s2 = block_reduce_sum(sv * sv, red, threadIdx.x, 128);
  if (threadIdx.x == 0) atomicAdd(&mincut_den[b], dflat[row] * s2);
}

// ---------------------------------------------------------------------------
// out[b,k,e] += sum_n s[b,n,k] * (X*mask)[b,n,e]   (LDS 100x128 accumulator)
// grid: (n_chunks=64, e_tiles=4, B)
// ---------------------------------------------------------------------------
__global__ __launch_bounds__(256) void pool_outer(
    const float* __restrict__ s, const float* __restrict__ X,
    const float* __restrict__ mask, float* __restrict__ outp)
{
  __shared__ float acc[100 * 128];
  __shared__ float srow[100];
  int b = blockIdx.z;
  int e0 = blockIdx.y * 128;
  int n0 = blockIdx.x * 128;
  int t = threadIdx.x;
  for (int i = t; i < 100 * 128; i += 256) acc[i] = 0.f;
  __syncthreads();
  int eloc = t & 127;
  int khalf = t >> 7;  // 0/1
  for (int n = n0; n < n0 + 128; n++) {
    long long row = (long long)b * 8192 + n;
    if (t < 100) srow[t] = s[row * 100 + t];
    __syncthreads();
    float xv = X[row * 512 + e0 + eloc] * mask[row];
    for (int k = khalf * 50; k < khalf * 50 + 50; k++)
      acc[k * 128 + eloc] += srow[k] * xv;
    __syncthreads();
  }
  for (int i = t; i < 100 * 128; i += 256) {
    int k = i >> 7, e = i & 127;
    atomicAdd(&outp[((long long)b * 100 + k) * 512 + e0 + e], acc[i]);
  }
}

// ---------------------------------------------------------------------------
// out_adj[b,k,l] += sum_n s[n,k]*adjs[n,l];  ss[b,k,l] += sum_n s[n,k]*s[n,l]
// grid: (n_chunks=16 of 512, kl_tiles=16 (4x4 of 32x32), B)
// ---------------------------------------------------------------------------
__global__ __launch_bounds__(256) void pool_gram(
    const float* __restrict__ s, const float* __restrict__ adjs,
    float* __restrict__ oadj, float* __restrict__ ssm)
{
  __shared__ float srow[128];
  __shared__ float arow[128];
  int b = blockIdx.z;
  int tile = blockIdx.y;
  int k0 = (tile >> 2) * 32, l0 = (tile & 3) * 32;
  int n0 = blockIdx.x * 512;
  int t = threadIdx.x;
  int kk = t >> 5;   // 0..7
  int ll = t & 31;   // 0..31
  float accA[4] = {0.f, 0.f, 0.f, 0.f};
  float accS[4] = {0.f, 0.f, 0.f, 0.f};
  for (int n = n0; n < n0 + 512; n++) {
    long long row = (long long)b * 8192 + n;
    if (t < 128) {
      srow[t] = (t < 100) ? s[row * 100 + t] : 0.f;
    } else {
      int i = t - 128;
      arow[i] = (i < 100) ? adjs[row * 100 + i] : 0.f;
    }
    __syncthreads();
    float al = arow[l0 + ll];
    float sl = srow[l0 + ll];
#pragma unroll
    for (int i = 0; i < 4; i++) {
      float sk = srow[k0 + kk + 8 * i];
      accA[i] += sk * al;
      accS[i] += sk * sl;
    }
    __syncthreads();
  }
#pragma unroll
  for (int i = 0; i < 4; i++) {
    int k = k0 + kk + 8 * i, l = l0 + ll;
    if (k < 100 && l < 100) {
      atomicAdd(&oadj[((long long)b * 100 + k) * 100 + l], accA[i]);
      atomicAdd(&ssm[((long long)b * 100 + k) * 100 + l], accS[i]);
    }
  }
}

// ---------------------------------------------------------------------------
// Per-batch: mincut term, ortho term, normalized out_adj -> d_out tail
// ---------------------------------------------------------------------------
__global__ __launch_bounds__(256) void finalize_batch(
    const float* __restrict__ oadj, const float* __restrict__ ssm,
    const float* __restrict__ mincut_den, float* __restrict__ batch_terms,
    float* __restrict__ oadj_out)
{
  __shared__ float Am[100 * 100];
  __shared__ float red[256];
  __shared__ float dvec[100];
  int b = blockIdx.x;
  int t = threadIdx.x;
  const float* oa = oadj + (long long)b * 10000;
  const float* sm = ssm + (long long)b * 10000;

  float tr = 0.f;
  for (int k = t; k < 100; k += 256) tr += oa[k * 100 + k];
  float num = block_reduce_sum(tr, red, t, 256);

  float f = 0.f;
  for (int i = t; i < 10000; i += 256) { float v = sm[i]; f += v * v; }
  float fro = sqrtf(block_reduce_sum(f, red, t, 256));
  float inv = 1.f / (fro + EPSC);
  float dk = 1.f / sqrtf(100.f);
  float o = 0.f;
  for (int i = t; i < 10000; i += 256) {
    int k = i / 100, l = i % 100;
    float v = sm[i] * inv - ((k == l) ? dk : 0.f);
    o += v * v;
  }
  float ortho = sqrtf(block_reduce_sum(o, red, t, 256));
  if (t == 0) {
    batch_terms[b * 2 + 0] = num / (mincut_den[b] + EPSC);
    batch_terms[b * 2 + 1] = ortho;
  }

  for (int i = t; i < 10000; i += 256) {
    int k = i / 100, l = i % 100;
    Am[i] = (k == l) ? 0.f : oa[i];
  }
  __syncthreads();
  for (int k = t; k < 100; k += 256) {
    float r = 0.f;
    for (int l = 0; l < 100; l++) r += Am[k * 100 + l];
    dvec[k] = sqrtf(r) + EPSC;
  }
  __syncthreads();
  for (int i = t; i < 10000; i += 256) {
    int k = i / 100, l = i % 100;
    oadj_out[(long long)b * 10000 + i] = Am[i] / (dvec[k] * dvec[l]);
  }
}

__global__ void finalize_loss(const float* __restrict__ bt, float* __restrict__ loss_out) {
  if (threadIdx.x == 0)
    loss_out[0] = -0.5f * (bt[0] + bt[2]) + 0.5f * (bt[1] + bt[3]);
}

// ---------------------------------------------------------------------------
// Attention + instance logits over Xp = [cls; pooled], grid = B*101
// ---------------------------------------------------------------------------
__global__ __launch_bounds__(256) void attn_pred(
    const float* __restrict__ outp, const float* __restrict__ cls_token,
    const float* __restrict__ aw1, const float* __restrict__ ab1,
    const float* __restrict__ aw2, const float* __restrict__ ab2,
    const float* __restrict__ pred_w, const float* __restrict__ pred_b,
    float* __restrict__ wlog)
{
  __shared__ float xp[512];
  __shared__ float red[256];
  int idx = blockIdx.x;
  int b = idx / 101, r = idx % 101;
  int t = threadIdx.x;
  const float* src = (r == 0) ? cls_token : (outp + ((long long)b * 100 + (r - 1)) * 512);
  for (int c = t; c < 512; c += 256) xp[c] = src[c];
  __syncthreads();

  float ap = 0.f;
  for (int j = t; j < 512; j += 256) {
    float h = ab1[j];
    for (int e = 0; e < 512; e++) h += xp[e] * aw1[e * 512 + j];
    ap += tanhf(h) * aw2[j];
  }
  float asum = block_reduce_sum(ap, red, t, 256);
  float attn = 1.f / (1.f + expf(-(asum + ab2[0])));

  float p0 = 0.f, p1 = 0.f;
  for (int e = t; e < 512; e += 256) {
    float x = xp[e];
    p0 += x * pred_w[e * 2 + 0];
    p1 += x * pred_w[e * 2 + 1];
  }
  float l0 = block_reduce_sum(p0, red, t, 256);
  float l1 = block_reduce_sum(p1, red, t, 256);
  if (t == 0) {
    wlog[idx * 2 + 0] = attn * (l0 + pred_b[0]);
    wlog[idx * 2 + 1] = attn * (l1 + pred_b[1]);
  }
}

// ---------------------------------------------------------------------------
// CrossGatingFusion: merged = LN(sig(pl@glw+glb)*pl + sig(pr@grw+grb)*pr)
// ---------------------------------------------------------------------------
__global__ __launch_bounds__(256) void fusion(
    const float* __restrict__ pl, const float* __restrict__ pr,
    const float* __restrict__ glw, const float* __restrict__ glb,
    const float* __restrict__ grw, const float* __restrict__ grb,
    const float* __restrict__ lng, const float* __restrict__ lnb,
    float* __restrict__ merged)
{
  __shared__ float lrow[1024];
  __shared__ float rrow[1024];
  __shared__ float mrow[1024];
  __shared__ float red[256];
  int p = blockIdx.x;
  int t = threadIdx.x;
  for (int c = t; c < 1024; c += 256) {
    lrow[c] = pl[(long long)p * 1024 + c];
    rrow[c] = pr[(long long)p * 1024 + c];
  }
  __syncthreads();
  for (int j = t; j < 1024; j += 256) {
    float gl = glb[j], gr = grb[j];
    for (int e = 0; e < 1024; e++) {
      gl += lrow[e] * glw[e * 1024 + j];
      gr += rrow[e] * grw[e * 1024 + j];
    }
    gl = 1.f / (1.f + expf(-gl));
    gr = 1.f / (1.f + expf(-gr));
    mrow[j] = gl * lrow[j] + gr * rrow[j];
  }
  __syncthreads();
  float s1 = 0.f;
  for (int j = t; j < 1024; j += 256) s1 += mrow[j];
  float mean = block_reduce_sum(s1, red, t, 256) * (1.f / 1024.f);
  float s2 = 0.f;
  for (int j = t; j < 1024; j += 256) { float d = mrow[j] - mean; s2 += d * d; }
  float var = block_reduce_sum(s2, red, t, 256) * (1.f / 1024.f);
  float inv = rsqrtf(var + 1e-5f);
  for (int j = t; j < 1024; j += 256)
    merged[(long long)p * 1024 + j] = (mrow[j] - mean) * inv * lng[j] + lnb[j];
}

// ---------------------------------------------------------------------------
extern "C" void kernel_launch(void* const* d_in, const int* in_sizes, int n_in,
                              void* d_out, int out_size, void* d_ws, size_t ws_size,
                              hipStream_t stream) {
  const float* node_feat = (const float*)d_in[0];
  const float* adj       = (const float*)d_in[1];
  const float* mask      = (const float*)d_in[2];
  const float* fc1_w     = (const float*)d_in[3];
  const float* fc1_b     = (const float*)d_in[4];
  const float* gcn_w     = (const float*)d_in[5];
  const float* gcn_b     = (const float*)d_in[6];
  const float* pool_w    = (const float*)d_in[7];
  const float* pool_b    = (const float*)d_in[8];
  const float* cls_tok   = (const float*)d_in[9];
  const float* aw1       = (const float*)d_in[10];
  const float* ab1       = (const float*)d_in[11];
  const float* aw2       = (const float*)d_in[12];
  const float* ab2       = (const float*)d_in[13];
  const float* pred_w    = (const float*)d_in[14];
  const float* pred_b    = (const float*)d_in[15];
  const float* gl_w      = (const float*)d_in[16];
  const float* gl_b      = (const float*)d_in[17];
  const float* gr_w      = (const float*)d_in[18];
  const float* gr_b      = (const float*)d_in[19];
  const float* ln_g      = (const float*)d_in[20];
  const float* ln_b      = (const float*)d_in[21];
  const float* protos_l  = (const float*)d_in[22];
  const float* protos_r  = (const float*)d_in[23];

  float* out = (float*)d_out;
  float* ws  = (float*)d_ws;

  // workspace layout (floats)
  const long long SZ_BNH = 16384LL * 512;   // 8,388,608
  float* X0    = ws;                         // fc1 output
  float* Y     = X0 + SZ_BNH;                // adj@X0 + X0
  float* Xn    = Y + SZ_BNH;                 // gcn output (normalized in place)
  float* S     = Xn + SZ_BNH;                // [2,8192,100]
  float* ADJS  = S + 16384LL * 100;          // [2,8192,100]
  float* DFLAT = ADJS + 16384LL * 100;       // [2,8192]
  float* OUTP  = DFLAT + 16384;              // [2,100,512] (atomic acc)
  float* OADJ  = OUTP + 2LL * 100 * 512;     // [2,100,100] (atomic acc)
  float* SSM   = OADJ + 2LL * 100 * 100;     // [2,100,100] (atomic acc)
  float* SCAL  = SSM + 2LL * 100 * 100;      // [0..1]=mincut_den, [2..5]=batch terms

  // zero the atomic accumulators + scalars (contiguous region)
  size_t zbytes = (size_t)(2LL * 100 * 512 + 2 * 2LL * 100 * 100 + 16) * sizeof(float);
  hipMemsetAsync(OUTP, 0, zbytes, stream);

  // d_out layout
  float* WLOG     = out;            // [2,101,2] = 404
  float* LOSS     = out + 404;      // scalar
  float* MERGED   = out + 405;      // [32,1024] = 32768
  float* OADJ_OUT = out + 33173;    // [2,100,100] = 20000

  // 1) X0 = relu(node_feat @ fc1_w + b) * mask   (M=16384,N=512,K=1024)
  wmma_gemm_f32<<<dim3((512 + BN - 1) / BN, 16384 / BM, 1), 256, 0, stream>>>(
      node_feat, fc1_w, fc1_b, nullptr, mask, X0, nullptr,
      16384, 512, 1024, 0, 0, 0, 1024, 512, 512, 1);

  // 2) Y = adj @ X0 + X0 ; d_flat = adj row sums (col-tile 0 only)
  wmma_gemm_f32<<<dim3((512 + BN - 1) / BN, 8192 / BM, 2), 256, 0, stream>>>(
      adj, X0, nullptr, X0, nullptr, Y, DFLAT,
      8192, 512, 8192, 8192LL * 8192, 8192LL * 512, 8192LL * 512,
      8192, 512, 512, 0);

  // 3) Xn = Y @ gcn_w + gcn_b
  wmma_gemm_f32<<<dim3((512 + BN - 1) / BN, 16384 / BM, 1), 256, 0, stream>>>(
      Y, gcn_w, gcn_b, nullptr, nullptr, Xn, nullptr,
      16384, 512, 512, 0, 0, 0, 512, 512, 512, 0);

  // 4) L2 normalize rows of Xn
  l2norm_rows<<<16384, 128, 0, stream>>>(Xn, 512);

  // 5) s = softmax(Xn @ pool_w + b) * mask ; mincut_den partials
  pool_softmax<<<16384, 128, 0, stream>>>(Xn, pool_w, pool_b, mask, DFLAT, S, SCAL);

  // 6) adj_s = adj @ s   (N=100, single col tile -> adj streamed exactly once)
  wmma_gemm_f32<<<dim3(1, 8192 / BM, 2), 256, 0, stream>>>(
      adj, S, nullptr, nullptr, nullptr, ADJS, nullptr,
      8192, 100, 8192, 8192LL * 8192, 8192LL * 100, 8192LL * 100,
      8192, 100, 100, 0);

  // 7) pooled features: OUTP = s^T (X*mask)
  pool_outer<<<dim3(64, 4, 2), 256, 0, stream>>>(S, Xn, mask, OUTP);

  // 8) gram: OADJ = s^T adj_s ; SSM = s^T s
  pool_gram<<<dim3(16, 16, 2), 256, 0, stream>>>(S, ADJS, OADJ, SSM);

  // 9) per-batch losses + normalized out_adj
  finalize_batch<<<2, 256, 0, stream>>>(OADJ, SSM, SCAL, SCAL + 2, OADJ_OUT);
  finalize_loss<<<1, 32, 0, stream>>>(SCAL + 2, LOSS);

  // 10) attention-weighted instance logits
  attn_pred<<<202, 256, 0, stream>>>(OUTP, cls_tok, aw1, ab1, aw2, ab2,
                                     pred_w, pred_b, WLOG);

  // 11) cross-gating fusion + layernorm
  fusion<<<32, 256, 0, stream>>>(protos_l, protos_r, gl_w, gl_b, gr_w, gr_b,
                                 ln_g, ln_b, MERGED);

  (void)in_sizes; (void)n_in; (void)out_size; (void)ws_size;
}